// GeomGCNSingleChannel_53669911331050
// MI455X (gfx1250) — compile-verified
//
#include <hip/hip_runtime.h>

typedef __attribute__((ext_vector_type(2))) float v2f;
typedef __attribute__((ext_vector_type(8))) float v8f;

#define INF   128          // in_feats
#define OUTF  64           // out_feats
#define DDIV  4            // divisions
#define HCOLS (DDIV * OUTF)   // 256 output cols per node
#define LDSW  (INF + 4)    // padded LDS row stride (132 floats): conflict-free b64 frags

// ---------------------------------------------------------------------------
// Kernel 1: per-division linear transform with f32 WMMA.
//   t[d][n][o] = norm[n] * sum_i feature[n][i] * W[d][o][i]
// One wave = 16 nodes x 64 outs for one division. W[d] staged in LDS.
// ---------------------------------------------------------------------------
__global__ __launch_bounds__(256) void geom_gemm(const float* __restrict__ feature,
                                                 const float* __restrict__ W,
                                                 const float* __restrict__ norm,
                                                 float* __restrict__ t,
                                                 int n_nodes) {
  __shared__ float ldsW[OUTF * LDSW];

  const int d    = blockIdx.y;
  const int wave = threadIdx.x >> 5;
  const int lane = threadIdx.x & 31;

  // Stage W[d] (64 x 128 f32 = 32 KB) into padded LDS.
  const float* Wd = W + (size_t)d * OUTF * INF;
  for (int i = threadIdx.x; i < OUTF * (INF / 4); i += 256) {
    const int row = i >> 5;            // i / 32  -> out row 0..63
    const int col = (i & 31) << 2;     // (i%32)*4 -> k col
    const float4 v = *(const float4*)(Wd + (size_t)row * INF + col);
    float* p = &ldsW[row * LDSW + col];
    p[0] = v.x; p[1] = v.y; p[2] = v.z; p[3] = v.w;
  }
  __syncthreads();

  const int tile = blockIdx.x * 8 + wave;        // 16-node tile index
  if (tile * 16 >= n_nodes) return;              // wave-uniform: EXEC stays all-1s
  const int node_base = tile * 16;

  const int r16   = lane & 15;                   // M (A) / N (B) index in tile
  const int khalf = (lane >> 4) << 1;            // 0 or 2: K pair selector

  const float* Arow = feature + (size_t)(node_base + r16) * INF + khalf;
  const float* B0 = &ldsW[(0 * 16 + r16) * LDSW + khalf];
  const float* B1 = &ldsW[(1 * 16 + r16) * LDSW + khalf];
  const float* B2 = &ldsW[(2 * 16 + r16) * LDSW + khalf];
  const float* B3 = &ldsW[(3 * 16 + r16) * LDSW + khalf];

  v8f acc0 = {}, acc1 = {}, acc2 = {}, acc3 = {};

#pragma unroll 8
  for (int k = 0; k < INF; k += 4) {
    const v2f a  = *(const v2f*)(Arow + k);   // global_load_b64
    const v2f b0 = *(const v2f*)(B0 + k);     // ds_load_b64 (bank-conflict-free)
    const v2f b1 = *(const v2f*)(B1 + k);
    const v2f b2 = *(const v2f*)(B2 + k);
    const v2f b3 = *(const v2f*)(B3 + k);
    acc0 = __builtin_amdgcn_wmma_f32_16x16x4_f32(false, a, false, b0, (short)0, acc0, false, false);
    acc1 = __builtin_amdgcn_wmma_f32_16x16x4_f32(false, a, false, b1, (short)0, acc1, false, false);
    acc2 = __builtin_amdgcn_wmma_f32_16x16x4_f32(false, a, false, b2, (short)0, acc2, false, false);
    acc3 = __builtin_amdgcn_wmma_f32_16x16x4_f32(false, a, false, b3, (short)0, acc3, false, false);
  }

  // C layout: VGPR r -> node = node_base + (lane/16)*8 + r ; col = ot*16 + (lane%16)
  const int mhalf = (lane >> 4) << 3;     // 0 or 8
  float nv[8];
#pragma unroll
  for (int r = 0; r < 8; ++r) nv[r] = norm[node_base + mhalf + r];
#pragma unroll
  for (int r = 0; r < 8; ++r) {
    acc0[r] *= nv[r]; acc1[r] *= nv[r]; acc2[r] *= nv[r]; acc3[r] *= nv[r];
  }

  float* tb = t + ((size_t)d * n_nodes + node_base + mhalf) * OUTF + r16;
#pragma unroll
  for (int r = 0; r < 8; ++r) {
    tb[(size_t)r * OUTF +  0] = acc0[r];
    tb[(size_t)r * OUTF + 16] = acc1[r];
    tb[(size_t)r * OUTF + 32] = acc2[r];
    tb[(size_t)r * OUTF + 48] = acc3[r];
  }
}

// ---------------------------------------------------------------------------
// Kernel 0: zero the output accumulator (float4 stores).
// ---------------------------------------------------------------------------
__global__ __launch_bounds__(256) void zero_out(float4* __restrict__ out, long long n4) {
  const long long i = (long long)blockIdx.x * 256 + threadIdx.x;
  if (i < n4) {
    float4 z; z.x = 0.f; z.y = 0.f; z.z = 0.f; z.w = 0.f;
    out[i] = z;
  }
}

// ---------------------------------------------------------------------------
// Kernel 2: edge gather + atomic scatter-sum.
// 16 threads per edge, 4 floats each; t rows are L2-resident (102 MB < 192 MB).
// out[dst][d*64 + o] += t[d][src][o]
// ---------------------------------------------------------------------------
__global__ __launch_bounds__(256) void edge_scatter(const int* __restrict__ src,
                                                    const int* __restrict__ dst,
                                                    const int* __restrict__ ediv,
                                                    const float* __restrict__ t,
                                                    float* __restrict__ out,
                                                    int n_edges, int n_nodes) {
  const long long gid = (long long)blockIdx.x * 256 + threadIdx.x;
  const long long e = gid >> 4;
  if (e >= n_edges) return;
  const int c = (int)(gid & 15);

  const int d = ediv[e];
  const int s = src[e];
  const int v = dst[e];

  const float4 m = *(const float4*)(t + ((size_t)d * n_nodes + s) * OUTF + c * 4);
  float* o = out + (size_t)v * HCOLS + d * OUTF + c * 4;
  atomicAdd(o + 0, m.x);
  atomicAdd(o + 1, m.y);
  atomicAdd(o + 2, m.z);
  atomicAdd(o + 3, m.w);
}

// ---------------------------------------------------------------------------
// Kernel 3: final dst-node norm scale + ReLU (in place).
// ---------------------------------------------------------------------------
__global__ __launch_bounds__(256) void finalize(const float* __restrict__ norm,
                                                float* __restrict__ out,
                                                long long total) {
  const long long i = (long long)blockIdx.x * 256 + threadIdx.x;
  if (i >= total) return;
  const int n = (int)(i >> 8);            // HCOLS == 256
  const float v = out[i] * norm[n];
  out[i] = fmaxf(v, 0.0f);
}

// ---------------------------------------------------------------------------
extern "C" void kernel_launch(void* const* d_in, const int* in_sizes, int n_in,
                              void* d_out, int out_size, void* d_ws, size_t ws_size,
                              hipStream_t stream) {
  const float* feature = (const float*)d_in[0];
  const float* W       = (const float*)d_in[1];
  const float* norm    = (const float*)d_in[2];
  const int*   src     = (const int*)d_in[3];
  const int*   dst     = (const int*)d_in[4];
  const int*   ediv    = (const int*)d_in[5];
  float* out = (float*)d_out;
  float* t   = (float*)d_ws;              // needs D*N*OUTF*4 = 102.4 MB

  const int n_nodes = in_sizes[2];        // norm has N elements
  const int n_edges = in_sizes[3];

  // 0: zero the accumulator (harness poisons d_out).
  const long long n4 = (long long)n_nodes * HCOLS / 4;
  zero_out<<<(unsigned)((n4 + 255) / 256), 256, 0, stream>>>((float4*)out, n4);

  // 1: WMMA GEMM -> t[d][n][o] (already scaled by source-node norm).
  const int tiles = (n_nodes + 15) / 16;
  dim3 g((tiles + 7) / 8, DDIV);
  geom_gemm<<<g, 256, 0, stream>>>(feature, W, norm, t, n_nodes);

  // 2: edge gather + atomic scatter.
  const long long ethreads = (long long)n_edges * 16;
  edge_scatter<<<(unsigned)((ethreads + 255) / 256), 256, 0, stream>>>(
      src, dst, ediv, t, out, n_edges, n_nodes);

  // 3: dst-node norm + ReLU.
  const long long ftotal = (long long)n_nodes * HCOLS;
  finalize<<<(unsigned)((ftotal + 255) / 256), 256, 0, stream>>>(norm, out, ftotal);
}